// MultiHeadAttentionLayer_77103252898070
// MI455X (gfx1250) — compile-verified
//
#include <hip/hip_runtime.h>
#include <hip/hip_bf16.h>
#include <math.h>

#define N_NODES 50000
#define N_EDGES 1600000
#define HIDDEN  128
#define HEADS   8
#define HEAD_DIM 16

typedef __attribute__((ext_vector_type(2))) float v2f;
typedef __attribute__((ext_vector_type(8))) float v8f;

// ---------------------------------------------------------------------------
// Kernel 1: degree accumulation (f32 atomics, L2-resident)
// ---------------------------------------------------------------------------
__global__ void deg_kernel(const int* __restrict__ src, const int* __restrict__ dst,
                           float* __restrict__ deg_out, float* __restrict__ deg_in) {
    int i = blockIdx.x * blockDim.x + threadIdx.x;
    if (i >= N_EDGES) return;
    atomicAdd(&deg_out[src[i]], 1.0f);
    atomicAdd(&deg_in[dst[i]], 1.0f);
}

// ---------------------------------------------------------------------------
// Kernel 2: norm = rsqrt(clip(deg,1))   (in place)
// ---------------------------------------------------------------------------
__global__ void norm_kernel(float* __restrict__ deg_out, float* __restrict__ deg_in) {
    int i = blockIdx.x * blockDim.x + threadIdx.x;
    if (i >= N_NODES) return;
    deg_out[i] = rsqrtf(fmaxf(deg_out[i], 1.0f));
    deg_in[i]  = rsqrtf(fmaxf(deg_in[i], 1.0f));
}

// ---------------------------------------------------------------------------
// Kernel 3: agg[dst] += h[src] * norm_out[src]   (one wave32 per edge,
// float4 per lane: 32 lanes x 4 floats = 128)
// ---------------------------------------------------------------------------
__global__ void edge_agg_kernel(const float* __restrict__ h,
                                const float* __restrict__ norm_out,
                                const int* __restrict__ src,
                                const int* __restrict__ dst,
                                float* __restrict__ agg) {
    int gid  = blockIdx.x * blockDim.x + threadIdx.x;
    int e    = gid >> 5;
    int lane = gid & 31;
    if (e >= N_EDGES) return;
    int s = src[e];
    int d = dst[e];
    float no = norm_out[s];
    float4 hv = reinterpret_cast<const float4*>(h + (size_t)s * HIDDEN)[lane];
    float* o = agg + (size_t)d * HIDDEN + lane * 4;
    atomicAdd(o + 0, hv.x * no);
    atomicAdd(o + 1, hv.y * no);
    atomicAdd(o + 2, hv.z * no);
    atomicAdd(o + 3, hv.w * no);
}

// ---------------------------------------------------------------------------
// Kernel 4: WMMA GEMM: out = relu((agg * norm_in[:,None]) @ W + b)
// f32 path via V_WMMA_F32_16X16X4_F32, 16x16 tile per wave, K=128 in steps
// of 4 (32 wmma ops/wave). W column panel (128x16) staged in LDS per block.
// Block = 128 threads = 4 waves (4 row-tiles), grid.y = 8 column tiles.
// ---------------------------------------------------------------------------
__global__ void __launch_bounds__(128)
gemm_wmma_kernel(const float* __restrict__ agg,
                 const float* __restrict__ norm_in,
                 const float* __restrict__ W,
                 const float* __restrict__ bias,
                 float* __restrict__ out) {
    __shared__ float ldsW[HIDDEN * 16];   // 128 rows (k) x 16 cols (n) = 8 KB

    const int ct   = blockIdx.y;          // column tile 0..7
    const int tid  = threadIdx.x;
    const int wave = tid >> 5;
    const int lane = tid & 31;

    // cooperative load of W[:, ct*16 : ct*16+16] into LDS, row-major [k][n]
    for (int i = tid; i < HIDDEN * 16; i += 128) {
        int k = i >> 4;
        int n = i & 15;
        ldsW[i] = W[k * HIDDEN + ct * 16 + n];
    }
    __syncthreads();

    const int rowTile = blockIdx.x * 4 + wave;        // 0..3124
    if (rowTile >= N_NODES / 16) return;
    const int rowBase = rowTile * 16;

    const int mLow   = lane & 15;                     // matrix row (A) / col (B)
    const int kShift = (lane < 16) ? 0 : 2;           // upper half-lanes take K+2

    const int   aRow  = rowBase + mLow;
    const float nin   = norm_in[aRow];
    const float* aPtr = agg + (size_t)aRow * HIDDEN;

    v8f acc = {};
    for (int k0 = 0; k0 < HIDDEN; k0 += 4) {
        v2f a, b;
        // A 16x4 fragment: lane m = lane%16, K = k0 + kShift + {0,1}
        a[0] = aPtr[k0 + kShift + 0] * nin;
        a[1] = aPtr[k0 + kShift + 1] * nin;
        // B 4x16 fragment (mirror of A): lane n = lane%16, K = k0 + kShift + {0,1}
        b[0] = ldsW[(k0 + kShift + 0) * 16 + mLow];
        b[1] = ldsW[(k0 + kShift + 1) * 16 + mLow];
        acc = __builtin_amdgcn_wmma_f32_16x16x4_f32(
            /*neg_a=*/false, a, /*neg_b=*/false, b,
            /*c_mod=*/(short)0, acc, /*reuse_a=*/false, /*reuse_b=*/false);
    }

    // epilogue: D layout -> lanes 0-15: M=v, lanes 16-31: M=v+8; N = lane%16
    const int   col  = ct * 16 + mLow;
    const float bcol = bias[col];
    const int   mOff = (lane < 16) ? 0 : 8;
#pragma unroll
    for (int v = 0; v < 8; ++v) {
        int row = rowBase + v + mOff;
        out[(size_t)row * HIDDEN + col] = fmaxf(acc[v] + bcol, 0.0f);
    }
}

// ---------------------------------------------------------------------------
// Kernel 5: per-edge attention. One wave32 per edge; lane owns 4 floats
// (head = lane/4). Head dot reduced over 4 lanes with wave32 shuffles.
// ---------------------------------------------------------------------------
__global__ void edge_attn_kernel(const float* __restrict__ Q,
                                 const float* __restrict__ K,
                                 const float* __restrict__ V,
                                 const int* __restrict__ src,
                                 const int* __restrict__ dst,
                                 float* __restrict__ wV,
                                 float* __restrict__ z) {
    int gid  = blockIdx.x * blockDim.x + threadIdx.x;
    int e    = gid >> 5;
    int lane = gid & 31;
    if (e >= N_EDGES) return;
    int s = src[e];
    int d = dst[e];

    float4 kv = reinterpret_cast<const float4*>(K + (size_t)s * HIDDEN)[lane];
    float4 qv = reinterpret_cast<const float4*>(Q + (size_t)d * HIDDEN)[lane];
    float partial = kv.x * qv.x + kv.y * qv.y + kv.z * qv.z + kv.w * qv.w;
    // reduce within head group of 4 lanes (wave32)
    partial += __shfl_xor(partial, 1, 32);
    partial += __shfl_xor(partial, 2, 32);

    // scale = sqrt(HEAD_DIM) = 4
    float sc = __expf(fminf(fmaxf(partial * 0.25f, -10.0f), 10.0f));

    float4 vv = reinterpret_cast<const float4*>(V + (size_t)s * HIDDEN)[lane];
    float* o = wV + (size_t)d * HIDDEN + lane * 4;
    atomicAdd(o + 0, vv.x * sc);
    atomicAdd(o + 1, vv.y * sc);
    atomicAdd(o + 2, vv.z * sc);
    atomicAdd(o + 3, vv.w * sc);
    if ((lane & 3) == 0) {
        atomicAdd(&z[(size_t)d * HEADS + (lane >> 2)], sc);
    }
}

// ---------------------------------------------------------------------------
// Kernel 6: out = wV / (z + 1e-6)   (in place on d_out)
// ---------------------------------------------------------------------------
__global__ void finalize_kernel(float* __restrict__ wV, const float* __restrict__ z) {
    int i = blockIdx.x * blockDim.x + threadIdx.x;
    if (i >= N_NODES * HIDDEN) return;
    int node = i / HIDDEN;
    int head = (i % HIDDEN) / HEAD_DIM;
    wV[i] = wV[i] / (z[(size_t)node * HEADS + head] + 1e-6f);
}

// ---------------------------------------------------------------------------
// Launch
// ---------------------------------------------------------------------------
extern "C" void kernel_launch(void* const* d_in, const int* in_sizes, int n_in,
                              void* d_out, int out_size, void* d_ws, size_t ws_size,
                              hipStream_t stream) {
    const float* h    = (const float*)d_in[0];
    const float* W_Q  = (const float*)d_in[1];
    const float* b_Q  = (const float*)d_in[2];
    const float* W_K  = (const float*)d_in[3];
    const float* b_K  = (const float*)d_in[4];
    const float* W_V  = (const float*)d_in[5];
    const float* b_V  = (const float*)d_in[6];
    const int*   src  = (const int*)d_in[7];
    const int*   dst  = (const int*)d_in[8];
    float*       outp = (float*)d_out;

    // workspace layout (floats)
    float* ws = (float*)d_ws;
    float* deg_out = ws;                                  // 50000   (becomes norm_out)
    float* deg_in  = ws + 50000;                          // 50000   (becomes norm_in)
    float* agg     = ws + 100000;                         // 6.4M
    float* zacc    = ws + 100000 + (size_t)N_NODES*HIDDEN;          // 400000
    float* Q       = zacc + (size_t)N_NODES*HEADS;                  // 6.4M
    float* Kf      = Q    + (size_t)N_NODES*HIDDEN;                 // 6.4M
    float* V       = Kf   + (size_t)N_NODES*HIDDEN;                 // 6.4M

    // zero: deg_out, deg_in, agg, z (contiguous prefix) and d_out (wV accum)
    size_t zero_prefix = (size_t)(100000 + N_NODES*HIDDEN + N_NODES*HEADS) * sizeof(float);
    hipMemsetAsync(d_ws, 0, zero_prefix, stream);
    hipMemsetAsync(d_out, 0, (size_t)N_NODES * HIDDEN * sizeof(float), stream);

    // 1) degrees
    deg_kernel<<<(N_EDGES + 255) / 256, 256, 0, stream>>>(src, dst, deg_out, deg_in);
    // 2) norms
    norm_kernel<<<(N_NODES + 255) / 256, 256, 0, stream>>>(deg_out, deg_in);
    // 3) shared aggregation (wave per edge)
    {
        long long waves = N_EDGES;
        long long blocks = (waves * 32 + 255) / 256;
        edge_agg_kernel<<<(int)blocks, 256, 0, stream>>>(h, deg_out, src, dst, agg);
    }
    // 4) three WMMA GEMMs: Q, K, V
    {
        dim3 grid((N_NODES / 16 + 3) / 4, HIDDEN / 16);   // (782, 8)
        gemm_wmma_kernel<<<grid, 128, 0, stream>>>(agg, deg_in, W_Q, b_Q, Q);
        gemm_wmma_kernel<<<grid, 128, 0, stream>>>(agg, deg_in, W_K, b_K, Kf);
        gemm_wmma_kernel<<<grid, 128, 0, stream>>>(agg, deg_in, W_V, b_V, V);
    }
    // 5) edge attention (wave per edge)
    {
        long long waves = N_EDGES;
        long long blocks = (waves * 32 + 255) / 256;
        edge_attn_kernel<<<(int)blocks, 256, 0, stream>>>(Q, Kf, V, src, dst, outp, zacc);
    }
    // 6) normalize
    finalize_kernel<<<(N_NODES * HIDDEN + 255) / 256, 256, 0, stream>>>(outp, zacc);
}